// GCN_24756191494148
// MI455X (gfx1250) — compile-verified
//
#include <hip/hip_runtime.h>
#include <cstdint>
#include <cstddef>

// ---------------------------------------------------------------------------
// Two-layer GCN (PyG GCNConv semantics) for MI455X / gfx1250, fp32 end-to-end.
// GEMMs use V_WMMA_F32_16X16X4_F32 (native f32 WMMA). Aggregation uses
// global_atomic_add_f32; all intermediates sized to stay L2-resident.
// ---------------------------------------------------------------------------

typedef __attribute__((ext_vector_type(2))) float v2f;
typedef __attribute__((ext_vector_type(8))) float v8f;

#define CH 64  // hidden/out channels (both layers)

// ---------------- elementwise helpers ----------------

__global__ void fill_f32(float* __restrict__ p, float v, size_t n) {
    size_t i = (size_t)blockIdx.x * blockDim.x + threadIdx.x;
    if (i < n) p[i] = v;
}

// deg[col[e]] += ew[e]  (deg pre-initialized to 1.0 for the self-loop)
__global__ void degree_accum(const long long* __restrict__ col,
                             const float* __restrict__ ew,
                             float* __restrict__ deg, long long E) {
    long long i = (long long)blockIdx.x * blockDim.x + threadIdx.x;
    if (i < E) atomicAdd(&deg[(size_t)col[i]], ew[i]);
}

// d[i] = deg>0 ? rsqrt(deg) : 0   (in place)
__global__ void deg_to_dinv(float* __restrict__ d, int n) {
    int i = blockIdx.x * blockDim.x + threadIdx.x;
    if (i < n) {
        float v = d[i];
        d[i] = (v > 0.f) ? rsqrtf(v) : 0.f;
    }
}

// ---------------- WMMA f32 GEMM:  C[N x 64] = A[N x K] * B[K x 64] ----------
// grid.x = ceil(N/16), block = 128 (4 waves). Wave w owns column tile w.
template<int K>
__global__ __launch_bounds__(128) void gemm_wmma_f32(const float* __restrict__ A,
                                                     const float* __restrict__ B,
                                                     float* __restrict__ C,
                                                     int nrows) {
    __shared__ float As[16 * (K + 4)];   // +4 pad: conflict-free column reads
    const int tid  = threadIdx.x;
    const int wave = tid >> 5;
    const int lane = tid & 31;
    const int row0 = blockIdx.x * 16;
    const int K4   = K / 4;

    // cooperative stage of the 16 x K A-tile (float4, fully coalesced)
    for (int idx = tid; idx < 16 * K4; idx += 128) {
        int r  = idx / K4;
        int c4 = idx - r * K4;
        if (row0 + r < nrows) {
            float4 v = ((const float4*)(A + (size_t)(row0 + r) * K))[c4];
            *(float4*)&As[r * (K + 4) + c4 * 4] = v;
        }
    }
    __syncthreads();

    // A frag (16x4 f32): lanes 0-15 -> K=0,1 ; lanes 16-31 -> K=2,3 (ISA layout)
    const int lm   = lane & 15;
    const int kh   = (lane >> 4) << 1;      // 0 or 2
    const int ncol = wave * 16 + lm;        // B/N column for this lane

    v8f acc = {};
    #pragma unroll
    for (int k = 0; k < K; k += 4) {
        v2f a, b;
        a.x = As[lm * (K + 4) + k + kh];
        a.y = As[lm * (K + 4) + k + kh + 1];
        b.x = B[(size_t)(k + kh)     * CH + ncol];
        b.y = B[(size_t)(k + kh + 1) * CH + ncol];
        // D = A*B + C : v_wmma_f32_16x16x4_f32
        acc = __builtin_amdgcn_wmma_f32_16x16x4_f32(
                  false, a, false, b, (short)0, acc, false, false);
    }

    // C/D layout: VGPR r -> (M=r, N=lane) for lanes 0-15, (M=8+r, N=lane-16) else
    const int m0 = (lane >> 4) << 3;        // 0 or 8
    float* cp = C + (size_t)row0 * CH + ncol;
    #pragma unroll
    for (int r = 0; r < 8; ++r) {
        if (row0 + m0 + r < nrows)
            cp[(size_t)(m0 + r) * CH] = acc[r];
    }
}

// ---------------- edge-wise scatter-add aggregation -------------------------
// One wave per edge: 32 lanes x float2 channels = 64 channels.
__global__ void edge_aggregate(const float* __restrict__ h,
                               const long long* __restrict__ row,
                               const long long* __restrict__ col,
                               const float* __restrict__ ew,
                               const float* __restrict__ dinv,
                               float* __restrict__ out, long long E) {
    long long t = (long long)blockIdx.x * blockDim.x + threadIdx.x;
    long long e = t >> 5;
    if (e >= E) return;
    int ci = (int)(t & 31) * 2;
    int r  = (int)row[e];
    int c  = (int)col[e];
    float norm = dinv[r] * ew[e] * dinv[c];      // wave-uniform
    const float2 hv = *(const float2*)(h + (size_t)r * CH + ci);
    float* op = out + (size_t)c * CH + ci;
    atomicAdd(op,     norm * hv.x);
    atomicAdd(op + 1, norm * hv.y);
}

// agg[i] = relu(agg[i] + dinv[node]^2 * hlin[i] + bias[c])   (self-loop fused)
__global__ void finalize_relu_inplace(float* __restrict__ agg,
                                      const float* __restrict__ hlin,
                                      const float* __restrict__ dinv,
                                      const float* __restrict__ bias,
                                      int n_nodes) {
    long long i = (long long)blockIdx.x * blockDim.x + threadIdx.x;
    if (i >= (long long)n_nodes * CH) return;
    int node = (int)(i >> 6);
    int c    = (int)(i & 63);
    float di = dinv[node];
    float v  = agg[i] + di * di * hlin[i] + bias[c];
    agg[i] = (v > 0.f) ? v : 0.f;
}

// outp[i] += dinv[node]^2 * hlin[i] + bias[c]   (no relu, layer 2)
__global__ void finalize_bias_inplace(float* __restrict__ outp,
                                      const float* __restrict__ hlin,
                                      const float* __restrict__ dinv,
                                      const float* __restrict__ bias,
                                      int n_nodes) {
    long long i = (long long)blockIdx.x * blockDim.x + threadIdx.x;
    if (i >= (long long)n_nodes * CH) return;
    int node = (int)(i >> 6);
    int c    = (int)(i & 63);
    float di = dinv[node];
    outp[i] = outp[i] + di * di * hlin[i] + bias[c];
}

// ---------------------------------------------------------------------------

extern "C" void kernel_launch(void* const* d_in, const int* in_sizes, int n_in,
                              void* d_out, int out_size, void* d_ws, size_t ws_size,
                              hipStream_t stream) {
    const float*     x  = (const float*)d_in[0];       // [N, 128]
    const long long* ei = (const long long*)d_in[1];   // [2, E] int64
    const float*     ew = (const float*)d_in[2];       // [E]
    const float*     W1 = (const float*)d_in[3];       // [128, 64]
    const float*     b1 = (const float*)d_in[4];       // [64]
    const float*     W2 = (const float*)d_in[5];       // [64, 64]
    const float*     b2 = (const float*)d_in[6];       // [64]
    float* out = (float*)d_out;                        // [N, 64]

    const int CIN = 128;
    const int N   = in_sizes[0] / CIN;                 // 100000
    const long long E = in_sizes[2];                   // 1600000

    const long long* row = ei;        // edge_index[0]
    const long long* col = ei + E;    // edge_index[1]

    // workspace layout (floats): dinv[N] | bufA[N*64] | bufB[N*64]  (~51.6 MB)
    float* dinv = (float*)d_ws;
    float* bufA = dinv + N;
    float* bufB = bufA + (size_t)N * CH;

    const int blk = 256;
    const size_t nfeat = (size_t)N * CH;
    const int gNode  = (N + blk - 1) / blk;
    const int gEdge  = (int)((E + blk - 1) / blk);
    const int gFeat  = (int)((nfeat + blk - 1) / blk);
    const int gEdgeW = (int)((E * 32 + blk - 1) / blk);
    const int gTile  = (N + 15) / 16;

    // ---- normalization: deg = 1 (self loop) + segment_sum(ew at col); dinv = rsqrt
    fill_f32<<<gNode, blk, 0, stream>>>(dinv, 1.0f, (size_t)N);
    degree_accum<<<gEdge, blk, 0, stream>>>(col, ew, dinv, E);
    deg_to_dinv<<<gNode, blk, 0, stream>>>(dinv, N);

    // ---- layer 1: bufA = x @ W1 ; bufB = scatter-add ; bufB = relu(+self+b1)
    gemm_wmma_f32<128><<<gTile, 128, 0, stream>>>(x, W1, bufA, N);
    fill_f32<<<gFeat, blk, 0, stream>>>(bufB, 0.0f, nfeat);
    edge_aggregate<<<gEdgeW, blk, 0, stream>>>(bufA, row, col, ew, dinv, bufB, E);
    finalize_relu_inplace<<<gFeat, blk, 0, stream>>>(bufB, bufA, dinv, b1, N);

    // ---- layer 2: bufA = bufB @ W2 ; out = scatter-add ; out += self + b2
    gemm_wmma_f32<64><<<gTile, 128, 0, stream>>>(bufB, W2, bufA, N);
    fill_f32<<<gFeat, blk, 0, stream>>>(out, 0.0f, nfeat);
    edge_aggregate<<<gEdgeW, blk, 0, stream>>>(bufA, row, col, ew, dinv, out, E);
    finalize_bias_inplace<<<gFeat, blk, 0, stream>>>(out, bufA, dinv, b2, N);
}